// RelationGraphConv_48009144434781
// MI455X (gfx1250) — compile-verified
//
#include <hip/hip_runtime.h>
#include <hip/hip_bf16.h>

#define N_SRC_C   50000
#define N_DST_C   50000
#define E_NUM_C   800000
#define D_IN_C    256
#define HD_C      256          // H*D
#define H_C       8
#define DH_C      32           // per-head dim
#define R_DIM_C   64
#define NEG_SLOPE_C 0.2f

typedef float v2f __attribute__((ext_vector_type(2)));
typedef float v8f __attribute__((ext_vector_type(8)));

// ---------------------------------------------------------------------------
// K0: rel = rel_emb[64] @ rel_w[64,512]
// ---------------------------------------------------------------------------
__global__ void rel_proj_kernel(const float* __restrict__ rel_emb,
                                const float* __restrict__ rel_w,
                                float* __restrict__ rel_out) {
  const int j = threadIdx.x;            // 0..511
  float acc = 0.f;
#pragma unroll 8
  for (int r = 0; r < R_DIM_C; ++r)
    acc += rel_emb[r] * rel_w[r * 512 + j];
  rel_out[j] = acc;
}

// ---------------------------------------------------------------------------
// K1: Y[50000,256] = X[50000,256] @ W[256,256] via V_WMMA_F32_16X16X4_F32
// One wave -> 16x64 output (4 accumulators, A fragment reused 4x).
// Block = 4 waves = one full 16-row x 256-col stripe.
// Fused: per-(row,head) attention logit e[n,h] = sum_d Y[n,h*32+d]*rel[h*64+off+d]
// computed in-register (each wave's 64 cols == exactly 2 heads).
// ---------------------------------------------------------------------------
__global__ void __launch_bounds__(128)
gemm_wmma_kernel(const float* __restrict__ X, const float* __restrict__ W,
                 float* __restrict__ Y,
                 const float* __restrict__ rel, int relOff,
                 float* __restrict__ e_out) {
  const int lane = threadIdx.x & 31;
  const int wave = threadIdx.x >> 5;          // 0..3
  const int m0 = blockIdx.x * 16;             // 3125 row tiles
  const int n0 = wave * 64;                   // 64-col group -> heads 2w, 2w+1

  // A fragment (16x4 f32): lane&15 = M row; (lane>>4)*2 = K pair offset.
  const int rA   = m0 + (lane & 15);
  const int koff = (lane >> 4) * 2;
  const int cL   = lane & 15;                 // within-tile column

  const float* xp = X + (size_t)rA * D_IN_C + koff;
  const float* wp = W + (size_t)koff * HD_C + n0 + cL;

  v8f c0 = {0.f,0.f,0.f,0.f,0.f,0.f,0.f,0.f};
  v8f c1 = c0, c2 = c0, c3 = c0;

#pragma unroll 2
  for (int k = 0; k < D_IN_C; k += 4) {
    v2f a, b0, b1, b2, b3;
    a.x = xp[0];  a.y = xp[1];
    b0.x = wp[0];        b0.y = wp[HD_C];
    b1.x = wp[16];       b1.y = wp[HD_C + 16];
    b2.x = wp[32];       b2.y = wp[HD_C + 32];
    b3.x = wp[48];       b3.y = wp[HD_C + 48];
    c0 = __builtin_amdgcn_wmma_f32_16x16x4_f32(false, a, false, b0, (short)0, c0, false, false);
    c1 = __builtin_amdgcn_wmma_f32_16x16x4_f32(false, a, false, b1, (short)0, c1, false, false);
    c2 = __builtin_amdgcn_wmma_f32_16x16x4_f32(false, a, false, b2, (short)0, c2, false, false);
    c3 = __builtin_amdgcn_wmma_f32_16x16x4_f32(false, a, false, b3, (short)0, c3, false, false);
    xp += 4;
    wp += 4 * HD_C;
  }

  // rel weights for this lane's 4 columns (cols n0+j*16+cL):
  // head(col) = col>>5, d(col) = col&31 ; rel row stride = 2*DH = 64
  const int head0 = (n0 >> 5);
  const int head1 = head0 + 1;
  const float w0 = rel[((n0 +  0 + cL) >> 5) * 64 + relOff + ((n0 +  0 + cL) & 31)];
  const float w1 = rel[((n0 + 16 + cL) >> 5) * 64 + relOff + ((n0 + 16 + cL) & 31)];
  const float w2 = rel[((n0 + 32 + cL) >> 5) * 64 + relOff + ((n0 + 32 + cL) & 31)];
  const float w3 = rel[((n0 + 48 + cL) >> 5) * 64 + relOff + ((n0 + 48 + cL) & 31)];

  // D layout: VGPR v -> row m0 + v + 8*(lane>=16), col n0 + j*16 + cL
  const int rowBase = m0 + (lane >> 4) * 8;
  float* yp = Y + (size_t)rowBase * HD_C + n0 + cL;

#pragma unroll
  for (int v = 0; v < 8; ++v) {
    yp[(size_t)v * HD_C +  0] = c0[v];
    yp[(size_t)v * HD_C + 16] = c1[v];
    yp[(size_t)v * HD_C + 32] = c2[v];
    yp[(size_t)v * HD_C + 48] = c3[v];

    // per-row logit partials for the two heads this wave owns
    float p0 = c0[v] * w0 + c1[v] * w1;   // head0 (cols j=0,1)
    float p1 = c2[v] * w2 + c3[v] * w3;   // head1 (cols j=2,3)
    // reduce across the 16 lanes holding this row (masks stay in half-wave)
    p0 += __shfl_xor(p0, 1); p0 += __shfl_xor(p0, 2);
    p0 += __shfl_xor(p0, 4); p0 += __shfl_xor(p0, 8);
    p1 += __shfl_xor(p1, 1); p1 += __shfl_xor(p1, 2);
    p1 += __shfl_xor(p1, 4); p1 += __shfl_xor(p1, 8);
    if (cL == 0) {
      const int row = rowBase + v;
      e_out[row * H_C + head0] = p0;
      e_out[row * H_C + head1] = p1;
    }
  }
}

// ---------------------------------------------------------------------------
// K3: zero-init m_enc / denom / out
// ---------------------------------------------------------------------------
__global__ void init_kernel(unsigned* __restrict__ m_enc,
                            float* __restrict__ denom,
                            float* __restrict__ out) {
  const int idx = blockIdx.x * blockDim.x + threadIdx.x;
  if (idx < N_DST_C * H_C) { m_enc[idx] = 0u; denom[idx] = 0.f; }
  if (idx < N_DST_C * HD_C) out[idx] = 0.f;
}

// order-preserving float -> uint encoding for atomic max
__device__ __forceinline__ unsigned enc_float(float f) {
  unsigned u = __float_as_uint(f);
  return (u & 0x80000000u) ? ~u : (u | 0x80000000u);
}

// ---------------------------------------------------------------------------
// K4: per (edge, head): e = leaky_relu(e_src[s]+e_dst[d]); atomic segment max
// ---------------------------------------------------------------------------
__global__ void edge_logit_max_kernel(const float* __restrict__ e_src,
                                      const float* __restrict__ e_dst,
                                      const int* __restrict__ src_idx,
                                      const int* __restrict__ dst_idx,
                                      float* __restrict__ e_edge,
                                      unsigned* __restrict__ m_enc) {
  const int idx = blockIdx.x * blockDim.x + threadIdx.x;
  if (idx >= E_NUM_C * H_C) return;
  const int ei = idx >> 3;
  const int h = idx & 7;
  const int s = src_idx[ei];
  const int d = dst_idx[ei];
  float t = e_src[s * H_C + h] + e_dst[d * H_C + h];
  t = (t > 0.f) ? t : NEG_SLOPE_C * t;
  e_edge[idx] = t;
  atomicMax(m_enc + d * H_C + h, enc_float(t));
}

// ---------------------------------------------------------------------------
// K5: decode m_enc -> float m in place; untouched / non-finite -> 0
// ---------------------------------------------------------------------------
__global__ void m_fixup_kernel(unsigned* __restrict__ m_enc) {
  const int idx = blockIdx.x * blockDim.x + threadIdx.x;
  if (idx >= N_DST_C * H_C) return;
  unsigned u = m_enc[idx];
  float m = 0.f;
  if (u != 0u) {
    unsigned orig = (u & 0x80000000u) ? (u & 0x7FFFFFFFu) : ~u;
    if ((orig & 0x7F800000u) != 0x7F800000u)  // finite check
      m = __uint_as_float(orig);
  }
  ((float*)m_enc)[idx] = m;
}

// ---------------------------------------------------------------------------
// K6: one wave32 per edge. ex = exp(e - m[d]); denom += ex (one lane per head);
//     out[d, :] += fs[s, :] * ex   (normalization deferred to K7)
// ---------------------------------------------------------------------------
__global__ void __launch_bounds__(256)
edge_scatter_kernel(const float* __restrict__ fs,
                    const float* __restrict__ e_edge,
                    const float* __restrict__ m,
                    float* __restrict__ denom,
                    const int* __restrict__ src_idx,
                    const int* __restrict__ dst_idx,
                    float* __restrict__ out) {
  const int lane = threadIdx.x & 31;
  const int ei = blockIdx.x * 8 + (threadIdx.x >> 5);
  if (ei >= E_NUM_C) return;
  const int s = src_idx[ei];
  const int d = dst_idx[ei];
  const int h = lane >> 2;  // 4 lanes per head (8 floats each, 32 per head)

  const float ex = __expf(e_edge[ei * H_C + h] - m[d * H_C + h]);
  if ((lane & 3) == 0) atomicAdd(denom + d * H_C + h, ex);

  const float4* fsp = (const float4*)(fs + (size_t)s * HD_C) + lane * 2;
  const float4 f0 = fsp[0];
  const float4 f1 = fsp[1];
  float* op = out + (size_t)d * HD_C + lane * 8;
  atomicAdd(op + 0, f0.x * ex);
  atomicAdd(op + 1, f0.y * ex);
  atomicAdd(op + 2, f0.z * ex);
  atomicAdd(op + 3, f0.w * ex);
  atomicAdd(op + 4, f1.x * ex);
  atomicAdd(op + 5, f1.y * ex);
  atomicAdd(op + 6, f1.z * ex);
  atomicAdd(op + 7, f1.w * ex);
}

// ---------------------------------------------------------------------------
// K7: out = relu(acc / denom)
// ---------------------------------------------------------------------------
__global__ void final_kernel(float* __restrict__ out,
                             const float* __restrict__ denom) {
  const int idx = blockIdx.x * blockDim.x + threadIdx.x;
  if (idx >= N_DST_C * HD_C) return;
  const int n = idx >> 8;
  const int h = (idx & 255) >> 5;
  const float den = denom[n * H_C + h];
  float v = out[idx];
  v = (den > 0.f) ? (v / den) : 0.f;
  out[idx] = (v > 0.f) ? v : 0.f;
}

// ---------------------------------------------------------------------------
extern "C" void kernel_launch(void* const* d_in, const int* in_sizes, int n_in,
                              void* d_out, int out_size, void* d_ws, size_t ws_size,
                              hipStream_t stream) {
  const float* feat_src = (const float*)d_in[0];
  const float* feat_dst = (const float*)d_in[1];
  const float* src_w    = (const float*)d_in[2];
  const float* dst_w    = (const float*)d_in[3];
  const float* rel_emb  = (const float*)d_in[4];
  const float* rel_w    = (const float*)d_in[5];
  const int*   src_idx  = (const int*)d_in[6];
  const int*   dst_idx  = (const int*)d_in[7];
  float* out = (float*)d_out;

  char* ws = (char*)d_ws;
  float* rel = (float*)ws;            ws += 4096;
  float* fs  = (float*)ws;            ws += (size_t)N_SRC_C * HD_C * 4;
  float* fd  = (float*)ws;            ws += (size_t)N_DST_C * HD_C * 4;
  float* e_src = (float*)ws;          ws += (size_t)N_SRC_C * H_C * 4;
  float* e_dst = (float*)ws;          ws += (size_t)N_DST_C * H_C * 4;
  unsigned* m_enc = (unsigned*)ws;    ws += (size_t)N_DST_C * H_C * 4;
  float* denom = (float*)ws;          ws += (size_t)N_DST_C * H_C * 4;
  float* e_edge = (float*)ws;         ws += (size_t)E_NUM_C * H_C * 4;

  rel_proj_kernel<<<1, 512, 0, stream>>>(rel_emb, rel_w, rel);

  // 3125 blocks x 4 waves: block = one 16-row x 256-col output stripe.
  // Fused attention-logit reduction: e_src uses rel[:, D:], e_dst rel[:, :D].
  gemm_wmma_kernel<<<3125, 128, 0, stream>>>(feat_src, src_w, fs, rel, DH_C, e_src);
  gemm_wmma_kernel<<<3125, 128, 0, stream>>>(feat_dst, dst_w, fd, rel, 0, e_dst);

  init_kernel<<<(N_DST_C * HD_C + 255) / 256, 256, 0, stream>>>(m_enc, denom, out);

  edge_logit_max_kernel<<<(E_NUM_C * H_C + 255) / 256, 256, 0, stream>>>(
      e_src, e_dst, src_idx, dst_idx, e_edge, m_enc);

  m_fixup_kernel<<<(N_DST_C * H_C + 255) / 256, 256, 0, stream>>>(m_enc);

  edge_scatter_kernel<<<E_NUM_C / 8, 256, 0, stream>>>(
      fs, e_edge, (const float*)m_enc, denom, src_idx, dst_idx, out);

  final_kernel<<<(N_DST_C * HD_C + 255) / 256, 256, 0, stream>>>(out, denom);
}